// Decoder_49374944035375
// MI455X (gfx1250) — compile-verified
//
#include <hip/hip_runtime.h>
#include <hip/hip_bf16.h>

// ---------------------------------------------------------------------------
// Attention-LSTM decoder for MI455X (gfx1250, wave32).
// GEMMs use V_WMMA_F32_16X16X4_F32 fed from a double-buffered LDS pipeline:
//   global(L2) -> regs (in-flight during WMMA) -> LDS -> ds_load fragments.
// Sizes: B=32, S=512, T=64, DENC=DHID=1024, DEMB=512, V=32000.
// ---------------------------------------------------------------------------

typedef float v8f __attribute__((ext_vector_type(8)));
typedef float v4f __attribute__((ext_vector_type(4)));
typedef float v2f __attribute__((ext_vector_type(2)));

#define Bsz   32
#define Ssz   512
#define Tsz   64
#define DENC  1024
#define DHID  1024
#define DEMB  512
#define VOC   32000

#define WMMA_F32(a, b, c) \
    __builtin_amdgcn_wmma_f32_16x16x4_f32(false, (a), false, (b), (short)0, (c), false, false)

// ---------------------------------------------------------------------------
// fp32 WMMA GEMM:  C[m,n] = A[m,:K] . W[n,:K] + bias[n] (+ C if acc)
// A: [32,lda] row-major (M fixed at 32 = batch), W: [N,ldw] row-major (A@W^T).
// Block = 4 waves -> 32(M) x 256(N) tile; per wave 32x64 (8 accumulators).
// K pipelined in chunks of 16 through double-buffered LDS.
// Grid: x = N/256. Block: (32,4). K % 16 == 0, N % 256 == 0 for all calls.
// ---------------------------------------------------------------------------
#define KC    16
#define BSTR  20   // padded LDS row stride (floats): conflict-free & 16B aligned

__global__ __launch_bounds__(128)
void gemm_wmma_f32(const float* __restrict__ A, int lda,
                   const float* __restrict__ W, int ldw,
                   const float* __restrict__ bias,
                   float* __restrict__ C, int ldc,
                   int N, int K, int accumulate)
{
    __shared__ float Bt[2][256 * BSTR];   // 2 x 20 KB
    __shared__ float At[2][32 * BSTR];    // 2 x 2.5 KB

    const int lane = threadIdx.x;                  // 0..31
    const int wv   = threadIdx.y;                  // 0..3
    const int tid  = wv * 32 + lane;               // 0..127
    const int nblk = blockIdx.x * 256;

    const int r    = lane & 15;                    // row within 16-tile
    const int koff = (lane >> 4) * 2;              // K sub-offset per half-wave

    // staging slots: B panel 256 rows x 16 floats = 1024 float4 / 128 thr = 8 each
    //                A panel  32 rows x 16 floats =  128 float4 / 128 thr = 1 each
    const int sArow = tid >> 2;
    const int sAq   = (tid & 3) * 4;

    v4f stB[8];
    v4f stA;

    auto issue_loads = [&](int kc) {
        #pragma unroll
        for (int i = 0; i < 8; ++i) {
            const int slot = tid + i * 128;
            const int row  = slot >> 2;
            const int q    = (slot & 3) * 4;
            stB[i] = *(const v4f*)(W + (size_t)(nblk + row) * ldw + kc + q);
        }
        stA = *(const v4f*)(A + (size_t)sArow * lda + kc + sAq);
    };
    auto store_stage = [&](int buf) {
        #pragma unroll
        for (int i = 0; i < 8; ++i) {
            const int slot = tid + i * 128;
            const int row  = slot >> 2;
            const int q    = (slot & 3) * 4;
            *(v4f*)&Bt[buf][row * BSTR + q] = stB[i];
        }
        *(v4f*)&At[buf][sArow * BSTR + sAq] = stA;
    };

    v8f c00 = {}, c01 = {}, c02 = {}, c03 = {};
    v8f c10 = {}, c11 = {}, c12 = {}, c13 = {};

    issue_loads(0);
    store_stage(0);
    __syncthreads();

    const int nch   = K / KC;
    const int brow0 = wv * 64;
    for (int c = 0; c < nch; ++c) {
        if (c + 1 < nch) issue_loads((c + 1) * KC);   // in flight during WMMAs

        const int buf = c & 1;
        const float* __restrict__ Bb = &Bt[buf][0];
        const float* __restrict__ Ab = &At[buf][0];
        #pragma unroll
        for (int kg = 0; kg < 4; ++kg) {
            const int kk = kg * 4 + koff;
            v2f a0 = *(const v2f*)&Ab[(r)      * BSTR + kk];
            v2f a1 = *(const v2f*)&Ab[(16 + r) * BSTR + kk];
            v2f b0 = *(const v2f*)&Bb[(brow0 +  0 + r) * BSTR + kk];
            v2f b1 = *(const v2f*)&Bb[(brow0 + 16 + r) * BSTR + kk];
            v2f b2 = *(const v2f*)&Bb[(brow0 + 32 + r) * BSTR + kk];
            v2f b3 = *(const v2f*)&Bb[(brow0 + 48 + r) * BSTR + kk];
            c00 = WMMA_F32(a0, b0, c00);
            c01 = WMMA_F32(a0, b1, c01);
            c02 = WMMA_F32(a0, b2, c02);
            c03 = WMMA_F32(a0, b3, c03);
            c10 = WMMA_F32(a1, b0, c10);
            c11 = WMMA_F32(a1, b1, c11);
            c12 = WMMA_F32(a1, b2, c12);
            c13 = WMMA_F32(a1, b3, c13);
        }
        if (c + 1 < nch) store_stage((c + 1) & 1);    // other buffer: safe
        __syncthreads();
    }

    // C/D layout: VGPR i -> M = i + (lane>>4)*8 ; N = n0 + (lane&15)
    const int n0  = nblk + wv * 64;
    const int col = lane & 15;
    const int rb  = (lane >> 4) * 8;
    const float bz = 0.0f;
    #pragma unroll
    for (int i = 0; i < 8; ++i) {
        int n; float v;
        {   // M rows 0..15
            float* __restrict__ crow = C + (size_t)(rb + i) * (size_t)ldc;
            n = n0 +  0 + col; v = c00[i] + (bias ? bias[n] : bz); if (accumulate) v += crow[n]; crow[n] = v;
            n = n0 + 16 + col; v = c01[i] + (bias ? bias[n] : bz); if (accumulate) v += crow[n]; crow[n] = v;
            n = n0 + 32 + col; v = c02[i] + (bias ? bias[n] : bz); if (accumulate) v += crow[n]; crow[n] = v;
            n = n0 + 48 + col; v = c03[i] + (bias ? bias[n] : bz); if (accumulate) v += crow[n]; crow[n] = v;
        }
        {   // M rows 16..31
            float* __restrict__ crow = C + (size_t)(16 + rb + i) * (size_t)ldc;
            n = n0 +  0 + col; v = c10[i] + (bias ? bias[n] : bz); if (accumulate) v += crow[n]; crow[n] = v;
            n = n0 + 16 + col; v = c11[i] + (bias ? bias[n] : bz); if (accumulate) v += crow[n]; crow[n] = v;
            n = n0 + 32 + col; v = c12[i] + (bias ? bias[n] : bz); if (accumulate) v += crow[n]; crow[n] = v;
            n = n0 + 48 + col; v = c13[i] + (bias ? bias[n] : bz); if (accumulate) v += crow[n]; crow[n] = v;
        }
    }
}

// ---------------------------------------------------------------------------
// Attention: one block per batch row. score -> softmax -> ctx.
// ---------------------------------------------------------------------------
__global__ __launch_bounds__(512)
void attn_ctx_kernel(const float* __restrict__ enc, const float* __restrict__ inp_mask,
                     const float* __restrict__ h, float* __restrict__ ctx)
{
    __shared__ float sh_h[DENC];
    __shared__ float sh_attn[Ssz];
    __shared__ float sred[Ssz];

    const int b   = blockIdx.x;
    const int tid = threadIdx.x;                          // 0..511
    const float* __restrict__ encb = enc + (size_t)b * Ssz * DENC;

    sh_h[tid]       = h[b * DHID + tid];
    sh_h[tid + 512] = h[b * DHID + tid + 512];
    __syncthreads();

    const float* __restrict__ row = encb + (size_t)tid * DENC;
    float s = 0.0f;
    for (int d = 0; d < DENC; d += 4) {
        s = fmaf(row[d+0], sh_h[d+0], s);
        s = fmaf(row[d+1], sh_h[d+1], s);
        s = fmaf(row[d+2], sh_h[d+2], s);
        s = fmaf(row[d+3], sh_h[d+3], s);
    }
    s -= (1.0f - inp_mask[b * Ssz + tid]) * 1e20f;

    sred[tid] = s; __syncthreads();
    for (int off = 256; off > 0; off >>= 1) {
        if (tid < off) sred[tid] = fmaxf(sred[tid], sred[tid + off]);
        __syncthreads();
    }
    const float mx = sred[0]; __syncthreads();
    const float ex = __expf(s - mx);
    sred[tid] = ex; __syncthreads();
    for (int off = 256; off > 0; off >>= 1) {
        if (tid < off) sred[tid] += sred[tid + off];
        __syncthreads();
    }
    const float denom = sred[0];
    sh_attn[tid] = ex / denom;
    __syncthreads();

    float a0 = 0.0f, a1 = 0.0f;
    for (int si = 0; si < Ssz; ++si) {
        const float w = sh_attn[si];
        const float* __restrict__ er = encb + (size_t)si * DENC;
        a0 = fmaf(w, er[tid],       a0);
        a1 = fmaf(w, er[tid + 512], a1);
    }
    ctx[b * DENC + tid]       = a0;
    ctx[b * DENC + tid + 512] = a1;
}

// ---------------------------------------------------------------------------
// Pointwise helpers
// ---------------------------------------------------------------------------
__global__ void init_state_kernel(const float* __restrict__ h0, const float* __restrict__ c0,
                                  const float* __restrict__ e0,
                                  float* __restrict__ h, float* __restrict__ c,
                                  float* __restrict__ e)
{
    const int idx = blockIdx.x * blockDim.x + threadIdx.x;    // 32*1024 threads
    const int b = idx >> 10, d = idx & 1023;
    h[idx] = h0[d];
    c[idx] = c0[d];
    if (d < DEMB) e[b * DEMB + d] = e0[d];
}

__global__ void concat2_kernel(const float* __restrict__ a, int na,
                               const float* __restrict__ bb, int nb,
                               float* __restrict__ out, int total)
{
    const int idx = blockIdx.x * blockDim.x + threadIdx.x;
    if (idx >= total) return;
    const int n = na + nb;
    const int b = idx / n, j = idx % n;
    out[idx] = (j < na) ? a[b * na + j] : bb[b * nb + (j - na)];
}

__global__ void gather_emb_kernel(const float* __restrict__ emb_w, const int* __restrict__ label,
                                  int t, float* __restrict__ e)
{
    const int b = blockIdx.x;
    const int d = threadIdx.x;                                // 512 threads
    int w = label[b * Tsz + t];
    w = (w < 0) ? 0 : (w >= VOC ? VOC - 1 : w);
    e[b * DEMB + d] = emb_w[(size_t)w * DEMB + d];
}

__device__ __forceinline__ float sigmoidf(float x) { return 1.0f / (1.0f + __expf(-x)); }

__global__ void lstm_cell_kernel(const float* __restrict__ gates,
                                 float* __restrict__ c, float* __restrict__ h)
{
    const int idx = blockIdx.x * blockDim.x + threadIdx.x;    // 32*1024
    const int b = idx >> 10, d = idx & 1023;
    const float* __restrict__ g = gates + (size_t)b * 4 * DHID;
    const float gi = g[d];
    const float gf = g[d + DHID];
    const float gg = g[d + 2 * DHID];
    const float go = g[d + 3 * DHID];
    const float cn = sigmoidf(gf) * c[idx] + sigmoidf(gi) * tanhf(gg);
    c[idx] = cn;
    h[idx] = sigmoidf(go) * tanhf(cn);
}

// ---------------------------------------------------------------------------
// Launch
// ---------------------------------------------------------------------------
extern "C" void kernel_launch(void* const* d_in, const int* in_sizes, int n_in,
                              void* d_out, int out_size, void* d_ws, size_t ws_size,
                              hipStream_t stream)
{
    const float* enc      = (const float*)d_in[0];
    const float* inp_mask = (const float*)d_in[1];
    const int*   label    = (const int*)  d_in[2];
    const float* emb_w    = (const float*)d_in[3];
    const float* W_ih     = (const float*)d_in[4];
    const float* b_ih     = (const float*)d_in[5];
    const float* W_hh     = (const float*)d_in[6];
    const float* b_hh     = (const float*)d_in[7];
    const float* proj_W   = (const float*)d_in[8];
    const float* proj_b   = (const float*)d_in[9];
    const float* emb0     = (const float*)d_in[10];
    const float* h0       = (const float*)d_in[11];
    const float* c0       = (const float*)d_in[12];
    float* out = (float*)d_out;

    // workspace carve-up (all sizes multiples of 64 floats)
    float* ws    = (float*)d_ws;
    float* h     = ws; ws += Bsz * DHID;            // 32x1024
    float* c     = ws; ws += Bsz * DHID;            // 32x1024
    float* e     = ws; ws += Bsz * DEMB;            // 32x512
    float* ctx   = ws; ws += Bsz * DENC;            // 32x1024
    float* xe    = ws; ws += Bsz * (DENC + DEMB);   // 32x1536
    float* gates = ws; ws += Bsz * 4 * DHID;        // 32x4096
    float* ch    = ws; ws += Bsz * (DENC + DHID);   // 32x2048
    float* p     = ws; ws += Bsz * DEMB;            // 32x512

    init_state_kernel<<<(Bsz * DHID) / 256, 256, 0, stream>>>(h0, c0, emb0, h, c, e);

    const dim3 gemm_block(32, 4);
    for (int t = 0; t < Tsz; ++t) {
        // attention over encoder states keyed by previous h
        attn_ctx_kernel<<<Bsz, 512, 0, stream>>>(enc, inp_mask, h, ctx);

        // x = [ctx, e]
        {
            const int total = Bsz * (DENC + DEMB);
            concat2_kernel<<<(total + 255) / 256, 256, 0, stream>>>(ctx, DENC, e, DEMB, xe, total);
        }
        // e for the NEXT step (teacher forcing); safe now that xe is built
        gather_emb_kernel<<<Bsz, DEMB, 0, stream>>>(emb_w, label, t, e);

        // gates = xe @ W_ih^T + b_ih ; gates += h @ W_hh^T + b_hh
        gemm_wmma_f32<<<dim3((4 * DHID) / 256), gemm_block, 0, stream>>>(
            xe, DENC + DEMB, W_ih, DENC + DEMB, b_ih, gates, 4 * DHID, 4 * DHID, DENC + DEMB, 0);
        gemm_wmma_f32<<<dim3((4 * DHID) / 256), gemm_block, 0, stream>>>(
            h, DHID, W_hh, DHID, b_hh, gates, 4 * DHID, 4 * DHID, DHID, 1);

        // LSTM cell: updates c, h in place
        lstm_cell_kernel<<<(Bsz * DHID) / 256, 256, 0, stream>>>(gates, c, h);

        // p = [ctx, h_new] @ proj_W^T + proj_b
        {
            const int total = Bsz * (DENC + DHID);
            concat2_kernel<<<(total + 255) / 256, 256, 0, stream>>>(ctx, DENC, h, DHID, ch, total);
        }
        gemm_wmma_f32<<<dim3(DEMB / 256), gemm_block, 0, stream>>>(
            ch, DENC + DHID, proj_W, DENC + DHID, proj_b, p, DEMB, DEMB, DENC + DHID, 0);

        // out[:, t, :] = p @ emb_w^T   (row stride in C is T*V)
        gemm_wmma_f32<<<dim3(VOC / 256), gemm_block, 0, stream>>>(
            p, DEMB, emb_w, DEMB, nullptr, out + (size_t)t * VOC, Tsz * VOC, VOC, DEMB, 0);
    }
}